// MambaSSMZonosBackbone_11115375362565
// MI455X (gfx1250) — compile-verified
//
#include <hip/hip_runtime.h>
#include <hip/hip_bf16.h>
#include <cstdint>

// ---- problem dims ----
#define DM    1024      // D_MODEL
#define DI    2048      // D_INNER
#define DSTT  16        // D_STATE
#define DCONV 4
#define DTR   64        // DT_RANK
#define NXDB  96        // DT_RANK + 2*D_STATE
#define NLAY  4
#define BATCH 2
#define SEQ   1024
#define NTOK  (BATCH*SEQ)   // 2048 tokens

typedef __attribute__((ext_vector_type(2))) float v2f;
typedef __attribute__((ext_vector_type(8))) float v8f;
typedef __attribute__((ext_vector_type(4))) unsigned int u32x4;
typedef __attribute__((ext_vector_type(8))) int i32x8;
typedef __attribute__((ext_vector_type(4))) int i32x4;

#ifndef __has_builtin
#define __has_builtin(x) 0
#endif
#if __has_builtin(__builtin_amdgcn_tensor_load_to_lds) && \
    __has_builtin(__builtin_amdgcn_s_wait_tensorcnt)
#define USE_TDM 1
#else
#define USE_TDM 0
#endif

// =====================================================================
// Fused residual-add + LayerNorm.  grid = NTOK blocks, 256 threads.
// =====================================================================
__global__ __launch_bounds__(256)
void add_norm_kernel(const float* __restrict__ hsrc, float* __restrict__ rbuf,
                     const float* __restrict__ w, const float* __restrict__ bvec,
                     float* __restrict__ out, int first)
{
    const int token = blockIdx.x;
    const int tid   = threadIdx.x;
    __shared__ float part[8];

    float vals[4];
    float s = 0.f;
    #pragma unroll
    for (int i = 0; i < 4; ++i) {
        int c = tid + i * 256;
        float v = hsrc[(size_t)token * DM + c];
        if (!first) v += rbuf[(size_t)token * DM + c];
        rbuf[(size_t)token * DM + c] = v;
        vals[i] = v;
        s += v;
    }
    #pragma unroll
    for (int o = 16; o > 0; o >>= 1) s += __shfl_xor(s, o, 32);
    if ((tid & 31) == 0) part[tid >> 5] = s;
    __syncthreads();
    float tot = 0.f;
    #pragma unroll
    for (int i = 0; i < 8; ++i) tot += part[i];
    const float mu = tot * (1.f / DM);
    __syncthreads();

    float vs = 0.f;
    #pragma unroll
    for (int i = 0; i < 4; ++i) { float d = vals[i] - mu; vs += d * d; }
    #pragma unroll
    for (int o = 16; o > 0; o >>= 1) vs += __shfl_xor(vs, o, 32);
    if ((tid & 31) == 0) part[tid >> 5] = vs;
    __syncthreads();
    float vtot = 0.f;
    #pragma unroll
    for (int i = 0; i < 8; ++i) vtot += part[i];
    const float inv = rsqrtf(vtot * (1.f / DM) + 1e-5f);

    #pragma unroll
    for (int i = 0; i < 4; ++i) {
        int c = tid + i * 256;
        out[(size_t)token * DM + c] = (vals[i] - mu) * inv * w[c] + bvec[c];
    }
}

// =====================================================================
// fp32 WMMA GEMM with TDM (tensor_load_to_lds) double-buffered staging.
//   C[M,N] = epi(A[M,K] * B[N,K]^T (+bias));  epi 1 = bias+softplus
// grid = (ceil(N/64), M/32), block = 128 (4 waves).
// Block tile: 32(M) x 64(N); wave w -> dual 16x16 tiles at n = n0b+16w.
// K staged 64 wide; LDS rows padded to 68 floats (68%64==4 -> no bank
// conflicts on the b64 fragment reads).  TDM hardware padding
// (pad_interval=64 DW, pad_amount=4 DW) reproduces the same stride.
// Pipeline: [wave0: s_wait_tensorcnt 0] -> barrier -> [wave0: issue TDM
// for chunk c+1 into buf^1] -> all waves WMMA on buf.  One barrier/iter.
// =====================================================================
#define BM  32
#define BN  64
#define KC  64
#define LDT 68

#if USE_TDM
__device__ __forceinline__ void tdm_load_2d(uint32_t lds_off, const void* gptr,
                                            uint32_t rem_x, uint32_t rem_y,
                                            uint32_t tile_x, uint32_t tile_y,
                                            uint32_t stride_x /*elems*/)
{
    uint64_t ga = (uint64_t)(uintptr_t)gptr;
    u32x4 g0;
    g0[0] = 1u;                                            // count=1, user D#
    g0[1] = lds_off;                                       // lds_addr (bytes)
    g0[2] = (uint32_t)ga;                                  // global_addr[31:0]
    g0[3] = (uint32_t)((ga >> 32) & 0x01FFFFFFu) | (2u << 30);  // ga[56:32] | type=2
    i32x8 g1;
    // data_size=2 (4B) | pad_enable | pad_interval=5 (64 DW) | pad_amount=3 (4 DW)
    g1[0] = (int)((2u << 16) | (1u << 20) | (5u << 22) | (3u << 25));
    g1[1] = (int)((rem_x & 0xFFFFu) << 16);                // tensor_dim0[15:0]
    g1[2] = (int)(((rem_x >> 16) & 0xFFFFu) | ((rem_y & 0xFFFFu) << 16));
    g1[3] = (int)(((rem_y >> 16) & 0xFFFFu) | (tile_x << 16));   // tile_dim0
    g1[4] = (int)(tile_y & 0xFFFFu);                       // tile_dim1 (tile_dim2=0)
    g1[5] = (int)stride_x;                                 // tensor_dim0_stride lo
    g1[6] = 0;                                             // stride hi / dim1_stride
    g1[7] = 0;
    i32x4 z4 = {0, 0, 0, 0};
    i32x8 z8 = {0, 0, 0, 0, 0, 0, 0, 0};
    __builtin_amdgcn_tensor_load_to_lds(g0, g1, z4, z4, z8, 0);
}
#endif

__global__ __launch_bounds__(128)
void gemm_f32_wmma(const float* __restrict__ A, int lda,
                   const float* __restrict__ Bw, int ldb,
                   float* __restrict__ C, int ldc,
                   int M, int N, int K,
                   const float* __restrict__ bias, int epi)
{
    __shared__ float sA[2][BM * LDT];
    __shared__ float sB[2][BN * LDT];

    const int tid  = threadIdx.x;
    const int wave = tid >> 5;
    const int lane = tid & 31;
    const int hf   = lane >> 4;
    const int rr   = lane & 15;
    const int m0   = blockIdx.y * BM;
    const int n0b  = blockIdx.x * BN;
    const int n0   = n0b + wave * 16;
    const int nchunks = K / KC;            // all call sites: K % 64 == 0

    v8f acc0 = {};
    v8f acc1 = {};

#if USE_TDM
    const int w0 = __builtin_amdgcn_readfirstlane(wave);   // scalar branch guard
    if (w0 == 0) {
        tdm_load_2d((uint32_t)(uintptr_t)&sA[0][0], &A[(size_t)m0 * lda],
                    (uint32_t)K, (uint32_t)(M - m0), KC, BM, (uint32_t)lda);
        tdm_load_2d((uint32_t)(uintptr_t)&sB[0][0], &Bw[(size_t)n0b * ldb],
                    (uint32_t)K, (uint32_t)(N - n0b), KC, BN, (uint32_t)ldb);
    }
#else
    {   // stage chunk 0 (float4-vectorized cooperative copy)
        #pragma unroll
        for (int i = 0; i < 4; ++i) {                      // 32x64 = 512 float4
            int e = (tid + i * 128) * 4;
            int row = e >> 6, col = e & 63;
            *(float4*)&sA[0][row * LDT + col] =
                *(const float4*)&A[(size_t)(m0 + row) * lda + col];
        }
        #pragma unroll
        for (int i = 0; i < 8; ++i) {                      // 64x64 = 1024 float4
            int e = (tid + i * 128) * 4;
            int row = e >> 6, col = e & 63;
            int nr = n0b + row; if (nr > N - 1) nr = N - 1;
            *(float4*)&sB[0][row * LDT + col] =
                *(const float4*)&Bw[(size_t)nr * ldb + col];
        }
    }
#endif

    for (int c = 0; c < nchunks; ++c) {
        const int cur = c & 1;
#if USE_TDM
        if (w0 == 0) __builtin_amdgcn_s_wait_tensorcnt(0); // chunk c landed
#endif
        __syncthreads();                                   // publish chunk c; buf^1 free
        if (c + 1 < nchunks) {
            const int kc = (c + 1) * KC;
#if USE_TDM
            if (w0 == 0) {
                tdm_load_2d((uint32_t)(uintptr_t)&sA[cur ^ 1][0],
                            &A[(size_t)m0 * lda + kc],
                            (uint32_t)(K - kc), (uint32_t)(M - m0), KC, BM,
                            (uint32_t)lda);
                tdm_load_2d((uint32_t)(uintptr_t)&sB[cur ^ 1][0],
                            &Bw[(size_t)n0b * ldb + kc],
                            (uint32_t)(K - kc), (uint32_t)(N - n0b), KC, BN,
                            (uint32_t)ldb);
            }
#else
            #pragma unroll
            for (int i = 0; i < 4; ++i) {
                int e = (tid + i * 128) * 4;
                int row = e >> 6, col = e & 63;
                *(float4*)&sA[cur ^ 1][row * LDT + col] =
                    *(const float4*)&A[(size_t)(m0 + row) * lda + kc + col];
            }
            #pragma unroll
            for (int i = 0; i < 8; ++i) {
                int e = (tid + i * 128) * 4;
                int row = e >> 6, col = e & 63;
                int nr = n0b + row; if (nr > N - 1) nr = N - 1;
                *(float4*)&sB[cur ^ 1][row * LDT + col] =
                    *(const float4*)&Bw[(size_t)nr * ldb + kc + col];
            }
#endif
        }
        if (n0 < N) {                                      // wave-uniform
            #pragma unroll
            for (int kk = 0; kk < KC; kk += 4) {
                v2f a0 = *(const v2f*)&sA[cur][rr * LDT + kk + 2 * hf];
                v2f a1 = *(const v2f*)&sA[cur][(16 + rr) * LDT + kk + 2 * hf];
                v2f b  = *(const v2f*)&sB[cur][(wave * 16 + rr) * LDT + kk + 2 * hf];
                acc0 = __builtin_amdgcn_wmma_f32_16x16x4_f32(
                           false, a0, false, b, (short)0, acc0, false, false);
                acc1 = __builtin_amdgcn_wmma_f32_16x16x4_f32(
                           false, a1, false, b, (short)0, acc1, false, false);
            }
        }
    }

    if (n0 < N) {
        const int n = n0 + rr;
        #pragma unroll
        for (int v = 0; v < 8; ++v) {
            int m = m0 + v + 8 * hf;
            float v0 = acc0[v];
            float v1 = acc1[v];
            if (epi == 1) {
                v0 += bias[n];
                v1 += bias[n];
                v0 = (v0 > 20.f) ? v0 : log1pf(__expf(v0));
                v1 = (v1 > 20.f) ? v1 : log1pf(__expf(v1));
            }
            C[(size_t)m * ldc + n] = v0;
            C[(size_t)(m + 16) * ldc + n] = v1;
        }
    }
}

// =====================================================================
// Depthwise causal conv (K=4) + SiLU over xz's first half.
// =====================================================================
__global__ __launch_bounds__(256)
void conv_silu_kernel(const float* __restrict__ xz, const float* __restrict__ w,
                      const float* __restrict__ bvec, float* __restrict__ xc)
{
    int idx = blockIdx.x * 256 + threadIdx.x;
    if (idx >= NTOK * DI) return;
    int c = idx & (DI - 1);
    int l = (idx / DI) & (SEQ - 1);
    int b = idx / (DI * SEQ);
    float acc = bvec[c];
    #pragma unroll
    for (int k = 0; k < DCONV; ++k) {
        int ll = l + k - (DCONV - 1);
        if (ll >= 0)
            acc += xz[((size_t)(b * SEQ + ll)) * (2 * DI) + c] * w[c * DCONV + k];
    }
    xc[idx] = acc / (1.f + __expf(-acc));    // silu
}

// =====================================================================
// Selective scan. thread = (channel d, state n): n = tid&15.
// =====================================================================
__global__ __launch_bounds__(256)
void scan_kernel(const float* __restrict__ xc, const float* __restrict__ dt,
                 const float* __restrict__ xdb, const float* __restrict__ A_log,
                 float* __restrict__ y)
{
    const int d = blockIdx.x * 16 + (threadIdx.x >> 4);
    const int n = threadIdx.x & 15;
    const int b = blockIdx.y;

    const float Adn = -__expf(A_log[(size_t)d * DSTT + n]);
    float h = 0.f;

    const float* dt_p = dt  + (size_t)b * SEQ * DI + d;
    const float* x_p  = xc  + (size_t)b * SEQ * DI + d;
    const float* B_p  = xdb + (size_t)b * SEQ * NXDB + DTR + n;
    const float* C_p  = xdb + (size_t)b * SEQ * NXDB + DTR + DSTT + n;
    float*       y_p  = y   + (size_t)b * SEQ * DI + d;

    for (int t = 0; t < SEQ; ++t) {
        float dtv = dt_p[(size_t)t * DI];
        float xv  = x_p [(size_t)t * DI];
        float Bv  = B_p [(size_t)t * NXDB];
        float Cv  = C_p [(size_t)t * NXDB];
        h = __expf(dtv * Adn) * h + (dtv * xv) * Bv;
        float p = h * Cv;
        p += __shfl_xor(p, 1, 32);
        p += __shfl_xor(p, 2, 32);
        p += __shfl_xor(p, 4, 32);
        p += __shfl_xor(p, 8, 32);
        if (n == 0) y_p[(size_t)t * DI] = p;
    }
}

// =====================================================================
// y = (y + xc * D_skip) * silu(z)
// =====================================================================
__global__ __launch_bounds__(256)
void gate_kernel(float* __restrict__ y, const float* __restrict__ xc,
                 const float* __restrict__ xz, const float* __restrict__ Dskip)
{
    int idx = blockIdx.x * 256 + threadIdx.x;
    if (idx >= NTOK * DI) return;
    int c = idx & (DI - 1);
    int token = idx / DI;
    float z = xz[(size_t)token * (2 * DI) + DI + c];
    float s = z / (1.f + __expf(-z));
    y[idx] = (y[idx] + xc[idx] * Dskip[c]) * s;
}

// =====================================================================
// Host orchestration
// =====================================================================
extern "C" void kernel_launch(void* const* d_in, const int* in_sizes, int n_in,
                              void* d_out, int out_size, void* d_ws, size_t ws_size,
                              hipStream_t stream)
{
    (void)in_sizes; (void)n_in; (void)out_size; (void)ws_size;

    const float* hidden    = (const float*)d_in[0];
    const float* norm_w    = (const float*)d_in[1];
    const float* norm_b    = (const float*)d_in[2];
    const float* in_proj_w = (const float*)d_in[3];
    const float* conv_w    = (const float*)d_in[4];
    const float* conv_b    = (const float*)d_in[5];
    const float* x_proj_w  = (const float*)d_in[6];
    const float* dt_proj_w = (const float*)d_in[7];
    const float* dt_proj_b = (const float*)d_in[8];
    const float* A_log     = (const float*)d_in[9];
    const float* D_skip    = (const float*)d_in[10];
    const float* out_pw    = (const float*)d_in[11];
    const float* normf_w   = (const float*)d_in[12];
    const float* normf_b   = (const float*)d_in[13];

    float* ws = (float*)d_ws;
    size_t off = 0;
    float* r_buf = ws + off; off += (size_t)NTOK * DM;
    float* x_buf = ws + off; off += (size_t)NTOK * DM;
    float* xz    = ws + off; off += (size_t)NTOK * 2 * DI;
    float* xc    = ws + off; off += (size_t)NTOK * DI;
    float* xdb   = ws + off; off += (size_t)NTOK * NXDB;
    float* dtb   = ws + off; off += (size_t)NTOK * DI;
    float* ybuf  = ws + off; off += (size_t)NTOK * DI;
    float* hbuf  = ws + off; off += (size_t)NTOK * DM;

    const int eltBlocks = (NTOK * DI + 255) / 256;

    for (int layer = 0; layer < NLAY; ++layer) {
        const float* hsrc = (layer == 0) ? hidden : hbuf;

        add_norm_kernel<<<NTOK, 256, 0, stream>>>(
            hsrc, r_buf, norm_w + (size_t)layer * DM, norm_b + (size_t)layer * DM,
            x_buf, layer == 0);

        // xz = x @ in_proj_w^T      (2048 x 4096, K=1024)
        gemm_f32_wmma<<<dim3(2 * DI / BN, NTOK / BM), 128, 0, stream>>>(
            x_buf, DM, in_proj_w + (size_t)layer * 2 * DI * DM, DM,
            xz, 2 * DI, NTOK, 2 * DI, DM, nullptr, 0);

        conv_silu_kernel<<<eltBlocks, 256, 0, stream>>>(
            xz, conv_w + (size_t)layer * DI * DCONV, conv_b + (size_t)layer * DI, xc);

        // xdb = xc @ x_proj_w^T     (2048 x 96, K=2048)
        gemm_f32_wmma<<<dim3((NXDB + BN - 1) / BN, NTOK / BM), 128, 0, stream>>>(
            xc, DI, x_proj_w + (size_t)layer * NXDB * DI, DI,
            xdb, NXDB, NTOK, NXDB, DI, nullptr, 0);

        // dt = softplus(xdb[:, :64] @ dt_proj_w^T + b)   (2048 x 2048, K=64)
        gemm_f32_wmma<<<dim3(DI / BN, NTOK / BM), 128, 0, stream>>>(
            xdb, NXDB, dt_proj_w + (size_t)layer * DI * DTR, DTR,
            dtb, DI, NTOK, DI, DTR, dt_proj_b + (size_t)layer * DI, 1);

        scan_kernel<<<dim3(DI / 16, BATCH), 256, 0, stream>>>(
            xc, dtb, xdb, A_log + (size_t)layer * DI * DSTT, ybuf);

        gate_kernel<<<eltBlocks, 256, 0, stream>>>(
            ybuf, xc, xz, D_skip + (size_t)layer * DI);

        // h = y_gated @ out_proj_w^T  (2048 x 1024, K=2048)
        gemm_f32_wmma<<<dim3(DM / BN, NTOK / BM), 128, 0, stream>>>(
            ybuf, DI, out_pw + (size_t)layer * DM * DI, DI,
            hbuf, DM, NTOK, DM, DI, nullptr, 0);
    }

    add_norm_kernel<<<NTOK, 256, 0, stream>>>(
        hbuf, r_buf, normf_w, normf_b, (float*)d_out, 0);
}